// VisionLSTM2_60215441490317
// MI455X (gfx1250) — compile-verified
//
#include <hip/hip_runtime.h>
#include <hip/hip_bf16.h>

#define DIMC   192
#define NBLK   12
#define INNER  384
#define DH     4
#define NH     96
#define PATCH  16
#define IMG    224
#define GRD    14
#define SEQ    196
#define BATCH  4
#define NCLS   1000
#define BS     (BATCH*SEQ)     // 784 = 49*16 = 7*7 tiles of 16
#define GIN_LD (3*INNER)       // 1152
#define KB     192             // K staging chunk (divides 192/384/1152)
#define GWAVES 7               // waves per GEMM workgroup (49 = 7*7 M tiles)
#define GTHREADS (GWAVES*32)   // 224

typedef __attribute__((ext_vector_type(2))) float v2f;
typedef __attribute__((ext_vector_type(8))) float v8f;

__device__ __forceinline__ float silu_f(float x) {
  return x / (1.0f + __expf(-x));
}
__device__ __forceinline__ float logsig_f(float x) {
  return (x >= 0.0f) ? -log1pf(__expf(-x)) : (x - log1pf(__expf(x)));
}

// ---------------------------------------------------------------------------
// Patch embedding: e = conv(x, patch_w, stride 16) -> + patch_b + pos
// ---------------------------------------------------------------------------
__global__ void patch_kernel(const float* __restrict__ x,
                             const float* __restrict__ pw,
                             const float* __restrict__ pb,
                             const float* __restrict__ pos,
                             float* __restrict__ h) {
  int idx = blockIdx.x * blockDim.x + threadIdx.x;
  if (idx >= BS * DIMC) return;
  int c = idx % DIMC;
  int s = (idx / DIMC) % SEQ;
  int b = idx / (DIMC * SEQ);
  int gy = s / GRD, gx = s % GRD;
  float acc = pb[c];
  for (int ci = 0; ci < 3; ++ci) {
    const float* xp = x + (((size_t)b * 3 + ci) * IMG + gy * PATCH) * IMG + gx * PATCH;
    const float* wp = pw + ((size_t)c * 3 + ci) * PATCH * PATCH;
    for (int py = 0; py < PATCH; ++py)
      for (int px = 0; px < PATCH; ++px)
        acc += xp[(size_t)py * IMG + px] * wp[py * PATCH + px];
  }
  acc += pos[((size_t)gy * GRD + gx) * DIMC + c];
  h[idx] = acc;
}

// ---------------------------------------------------------------------------
// LayerNorm over last dim D with (1+w) scale; optional sequence flip on input.
// One wave per row, wave32 shuffle reduction.
// ---------------------------------------------------------------------------
__global__ void ln_kernel(const float* __restrict__ in,
                          const float* __restrict__ w,
                          const float* __restrict__ bb,
                          float* __restrict__ out,
                          int D, int flip, float eps) {
  int row = blockIdx.x;                  // 0..BS-1
  int b = row / SEQ, s = row % SEQ;
  int sin = flip ? (SEQ - 1 - s) : s;
  const float* pin = in + ((size_t)(b * SEQ + sin)) * D;
  float sum = 0.f, sum2 = 0.f;
  for (int d = threadIdx.x; d < D; d += 32) {
    float v = pin[d];
    sum += v; sum2 += v * v;
  }
  for (int off = 16; off; off >>= 1) {
    sum  += __shfl_xor(sum,  off, 32);
    sum2 += __shfl_xor(sum2, off, 32);
  }
  float mu  = sum  / (float)D;
  float var = sum2 / (float)D - mu * mu;
  float r   = rsqrtf(var + eps);
  float* pout = out + (size_t)row * D;
  for (int d = threadIdx.x; d < D; d += 32)
    pout[d] = (pin[d] - mu) * r * (1.0f + w[d]) + bb[d];
}

// ---------------------------------------------------------------------------
// fp32 WMMA GEMM with async LDS staging of the weight tile.
//   C[M,N] = A[M,K] @ W[N,K]^T + bias[N]
// Workgroup = 7 waves; grid = (7, N/16). All 7 waves share one 16-column
// N-tile; wave w owns M-tile (blockIdx.x*7 + w). The 16xKB weight slab is
// staged into LDS once per workgroup per K-chunk via
// GLOBAL_LOAD_ASYNC_TO_LDS_B128 (ASYNCcnt path, no VGPR round trip), then
// consumed by all waves as ds_load_b64 fragments feeding
// V_WMMA_F32_16X16X4_F32. M (=49*16) and N (768/96/192) are multiples of 16,
// so EXEC is all-1s around every WMMA.
// ---------------------------------------------------------------------------
__global__ void wmma_gemm_lds(const float* __restrict__ A,
                              const float* __restrict__ W,
                              const float* __restrict__ bias,
                              float* __restrict__ C,
                              int N, int K, int lda, int ldc) {
  __shared__ float wtile[16 * KB];      // 12 KB static LDS
  int nt   = blockIdx.y;
  int tid  = threadIdx.x;               // 0..223
  int wave = tid >> 5;
  int lane = tid & 31;
  int half = lane >> 4;                 // 0 -> K 0,1 ; 1 -> K 2,3
  int l15  = lane & 15;
  int mt   = blockIdx.x * GWAVES + wave;
  int m    = mt * 16 + l15;
  int n    = nt * 16 + l15;

  const float* pA = A + (size_t)m * lda + half * 2;
  const float* wbase = W + (size_t)(nt * 16) * K;

  v8f acc;
  float bv = bias ? bias[n] : 0.0f;
#pragma unroll
  for (int r = 0; r < 8; ++r) acc[r] = bv;

  for (int k0 = 0; k0 < K; k0 += KB) {
    // ---- cooperative async stage of W[nt*16 .. +16)[k0 .. k0+KB) ----
    // 16 rows * KB/4 float4 chunks = 768 chunks over 224 threads
    for (int c = tid; c < 16 * (KB / 4); c += GTHREADS) {
      int row = c / (KB / 4);
      int kc  = (c % (KB / 4)) << 2;
      const float* gsrc = wbase + (size_t)row * K + k0 + kc;
      unsigned ldst = (unsigned)(uintptr_t)(wtile + row * KB + kc);
      asm volatile("global_load_async_to_lds_b128 %0, %1, off"
                   :: "v"(ldst), "v"(gsrc) : "memory");
    }
    asm volatile("s_wait_asynccnt 0x0" ::: "memory");
    __syncthreads();

    // ---- consume: 48 WMMA K-steps from LDS (B) + global (A) ----
    const float* pB = wtile + l15 * KB + half * 2;   // LDS fragment base
#pragma unroll 8
    for (int k = 0; k < KB; k += 4) {
      v2f a  = *(const v2f*)(pA + k0 + k);  // global_load_b64
      v2f bq = *(const v2f*)(pB + k);       // ds_load_b64
      acc = __builtin_amdgcn_wmma_f32_16x16x4_f32(
          /*neg_a=*/false, a, /*neg_b=*/false, bq,
          /*c_mod=*/(short)0, acc, /*reuse_a=*/false, /*reuse_b=*/false);
    }
    __syncthreads();                      // protect wtile before next stage
  }

  // C/D layout: VGPR r holds row (mt*16 + r + half*8), col n
  int mrow = mt * 16 + half * 8;
#pragma unroll
  for (int r = 0; r < 8; ++r)
    C[(size_t)(mrow + r) * ldc + n] = acc[r];
}

// ---------------------------------------------------------------------------
// Depthwise 3x3 conv over 14x14 grid on xm (= xi[..., :INNER], row stride 768)
// + bias + SiLU -> xa.
// ---------------------------------------------------------------------------
__global__ void dwconv_silu_kernel(const float* __restrict__ xi,
                                   const float* __restrict__ cw,
                                   const float* __restrict__ cb,
                                   float* __restrict__ xa) {
  int idx = blockIdx.x * blockDim.x + threadIdx.x;
  if (idx >= BS * INNER) return;
  int c = idx % INNER;
  int s = (idx / INNER) % SEQ;
  int b = idx / (INNER * SEQ);
  int gy = s / GRD, gx = s % GRD;
  float acc = 0.f;
  for (int dy = -1; dy <= 1; ++dy) {
    int yy = gy + dy;
    if (yy < 0 || yy >= GRD) continue;
    for (int dx = -1; dx <= 1; ++dx) {
      int xx = gx + dx;
      if (xx < 0 || xx >= GRD) continue;
      acc += xi[((size_t)(b * SEQ + yy * GRD + xx)) * (2 * INNER) + c] *
             cw[c * 9 + (dy + 1) * 3 + (dx + 1)];
    }
  }
  acc += cb[c];
  xa[idx] = silu_f(acc);
}

// ---------------------------------------------------------------------------
// Headwise 4x4 projections: q,k from xa ; v from xm. Writes gin = [q|k|v].
// ---------------------------------------------------------------------------
__global__ void qkv_kernel(const float* __restrict__ xa,
                           const float* __restrict__ xi,
                           const float* __restrict__ qw,
                           const float* __restrict__ kw,
                           const float* __restrict__ vw,
                           float* __restrict__ gin) {
  int idx = blockIdx.x * blockDim.x + threadIdx.x;
  if (idx >= BS * NH * DH) return;
  int o = idx % DH;
  int n = (idx / DH) % NH;
  int s = (idx / (DH * NH)) % SEQ;
  int b = idx / (DH * NH * SEQ);
  const float* xav = xa + ((size_t)(b * SEQ + s)) * INNER + n * DH;
  const float* xmv = xi + ((size_t)(b * SEQ + s)) * (2 * INNER) + n * DH;
  const float* qwr = qw + ((size_t)n * DH + o) * DH;
  const float* kwr = kw + ((size_t)n * DH + o) * DH;
  const float* vwr = vw + ((size_t)n * DH + o) * DH;
  float q = 0.f, k = 0.f, v = 0.f;
#pragma unroll
  for (int d = 0; d < DH; ++d) {
    q += xav[d] * qwr[d];
    k += xav[d] * kwr[d];
    v += xmv[d] * vwr[d];
  }
  size_t base = ((size_t)(b * SEQ + s)) * GIN_LD + n * DH + o;
  gin[base]             = q;
  gin[base + INNER]     = k;
  gin[base + 2 * INNER] = v;
}

// ---------------------------------------------------------------------------
// mLSTM: one 256-thread workgroup per (b, head); LDS-staged recurrence with
// fused per-head outnorm.
// ---------------------------------------------------------------------------
__global__ void mlstm_kernel(const float* __restrict__ gin,
                             const float* __restrict__ igA,
                             const float* __restrict__ fgA,
                             const float* __restrict__ onw,
                             const float* __restrict__ onb,
                             float* __restrict__ hout) {
  int inst = blockIdx.x;
  int b = inst / NH, n = inst % NH;
  __shared__ float qs[SEQ * DH], ks[SEQ * DH], vs[SEQ * DH];
  __shared__ float cs[SEQ], igs[SEQ];
  int tid = threadIdx.x;

  for (int i = tid; i < SEQ * DH; i += blockDim.x) {
    int s = i / DH, d = i % DH;
    size_t base = ((size_t)(b * SEQ + s)) * GIN_LD + n * DH + d;
    qs[i] = gin[base];
    ks[i] = gin[base + INNER];
    vs[i] = gin[base + 2 * INNER];
  }
  for (int s = tid; s < SEQ; s += blockDim.x) {
    size_t gi = ((size_t)(b * SEQ + s)) * NH + n;
    igs[s] = igA[gi];
    cs[s]  = logsig_f(fgA[gi]);
  }
  __syncthreads();
  if (tid == 0) {
    float acc = 0.f;
    for (int s = 0; s < SEQ; ++s) { acc += cs[s]; cs[s] = acc; }
  }
  __syncthreads();

  if (tid < SEQ) {
    int s = tid;
    float cS = cs[s];
    float m = -3.0e38f;
    for (int t = 0; t <= s; ++t)
      m = fmaxf(m, cS - cs[t] + igs[t]);
    float q0 = qs[s*4], q1 = qs[s*4+1], q2 = qs[s*4+2], q3 = qs[s*4+3];
    float sumC = 0.f, h0 = 0.f, h1 = 0.f, h2 = 0.f, h3 = 0.f;
    for (int t = 0; t <= s; ++t) {
      float dcoef = __expf(cS - cs[t] + igs[t] - m);
      float qk = (q0*ks[t*4] + q1*ks[t*4+1] + q2*ks[t*4+2] + q3*ks[t*4+3]) * 0.5f;
      float Cst = qk * dcoef;
      sumC += Cst;
      h0 += Cst * vs[t*4];   h1 += Cst * vs[t*4+1];
      h2 += Cst * vs[t*4+2]; h3 += Cst * vs[t*4+3];
    }
    float norm = fmaxf(fabsf(sumC), __expf(-m));
    float sc = 1.0f / (norm + 1e-6f);
    h0 *= sc; h1 *= sc; h2 *= sc; h3 *= sc;
    float mu = 0.25f * (h0 + h1 + h2 + h3);
    float e0 = h0 - mu, e1 = h1 - mu, e2 = h2 - mu, e3 = h3 - mu;
    float var = 0.25f * (e0*e0 + e1*e1 + e2*e2 + e3*e3);
    float r = rsqrtf(var + 1e-5f);
    size_t ob = ((size_t)(b * SEQ + s)) * INNER + n * DH;
    hout[ob + 0] = e0 * r * (1.0f + onw[n*DH+0]) + onb[n*DH+0];
    hout[ob + 1] = e1 * r * (1.0f + onw[n*DH+1]) + onb[n*DH+1];
    hout[ob + 2] = e2 * r * (1.0f + onw[n*DH+2]) + onb[n*DH+2];
    hout[ob + 3] = e3 * r * (1.0f + onw[n*DH+3]) + onb[n*DH+3];
  }
}

// ---------------------------------------------------------------------------
// h = (h + skip*xa) * silu(z), z = xi[..., INNER:2*INNER]
// ---------------------------------------------------------------------------
__global__ void gate_kernel(float* __restrict__ hg,
                            const float* __restrict__ xa,
                            const float* __restrict__ xi,
                            const float* __restrict__ skip) {
  int idx = blockIdx.x * blockDim.x + threadIdx.x;
  if (idx >= BS * INNER) return;
  int c = idx % INNER;
  size_t row = (size_t)(idx / INNER);
  float z = xi[row * (2 * INNER) + INNER + c];
  hg[idx] = (hg[idx] + skip[c] * xa[idx]) * silu_f(z);
}

// ---------------------------------------------------------------------------
// Residual add with optional sequence flip of the layer output.
// ---------------------------------------------------------------------------
__global__ void resadd_kernel(float* __restrict__ h,
                              const float* __restrict__ tmp, int flip) {
  int idx = blockIdx.x * blockDim.x + threadIdx.x;
  if (idx >= BS * DIMC) return;
  int d = idx % DIMC;
  int s = (idx / DIMC) % SEQ;
  int b = idx / (DIMC * SEQ);
  int so = flip ? (SEQ - 1 - s) : s;
  h[((size_t)(b * SEQ + so)) * DIMC + d] += tmp[idx];
}

// ---------------------------------------------------------------------------
// Head: pooled = [xn[:,0], xn[:,-1]] @ head_w.T + head_b  (M=4, thin -> VALU)
// ---------------------------------------------------------------------------
__global__ void head_kernel(const float* __restrict__ xn,
                            const float* __restrict__ hw,
                            const float* __restrict__ hb,
                            float* __restrict__ out) {
  int idx = blockIdx.x * blockDim.x + threadIdx.x;
  if (idx >= BATCH * NCLS) return;
  int nc = idx % NCLS;
  int b  = idx / NCLS;
  const float* r0 = xn + ((size_t)(b * SEQ + 0)) * DIMC;
  const float* r1 = xn + ((size_t)(b * SEQ + SEQ - 1)) * DIMC;
  const float* wr = hw + (size_t)nc * (2 * DIMC);
  float acc = hb[nc];
  for (int d = 0; d < DIMC; ++d)
    acc += r0[d] * wr[d] + r1[d] * wr[DIMC + d];
  out[idx] = acc;
}

// ---------------------------------------------------------------------------
extern "C" void kernel_launch(void* const* d_in, const int* in_sizes, int n_in,
                              void* d_out, int out_size, void* d_ws, size_t ws_size,
                              hipStream_t stream) {
  const float* x        = (const float*)d_in[0];
  const float* patch_w  = (const float*)d_in[1];
  const float* patch_b  = (const float*)d_in[2];
  const float* pos      = (const float*)d_in[3];
  const float* final_nw = (const float*)d_in[4];
  const float* final_nb = (const float*)d_in[5];
  const float* head_w   = (const float*)d_in[6];
  const float* head_b   = (const float*)d_in[7];
  // block i, param j: 0 norm_w,1 norm_b,2 proj_up_w,3 proj_up_b,4 q_w,5 k_w,
  // 6 v_w,7 conv_w,8 conv_b,9 ig_w,10 ig_b,11 fg_w,12 fg_b,13 outnorm_w,
  // 14 outnorm_b,15 skip,16 proj_down_w,17 proj_down_b

  float* ws  = (float*)d_ws;
  float* h    = ws;                          // BS*DIMC
  float* xn   = h    + BS * DIMC;            // BS*DIMC
  float* xi   = xn   + BS * DIMC;            // BS*768
  float* xa   = xi   + BS * 2 * INNER;       // BS*INNER
  float* gin  = xa   + BS * INNER;           // BS*1152
  float* igb  = gin  + BS * GIN_LD;          // BS*NH
  float* fgb  = igb  + BS * NH;              // BS*NH
  float* hout = fgb  + BS * NH;              // BS*INNER
  float* tmp  = hout + BS * INNER;           // BS*DIMC

  const int T = 256;
  patch_kernel<<<(BS * DIMC + T - 1) / T, T, 0, stream>>>(x, patch_w, patch_b, pos, h);

  for (int i = 0; i < NBLK; ++i) {
    auto P = [&](int j) { return (const float*)d_in[8 + i * 18 + j]; };
    int flip = i & 1;

    ln_kernel<<<BS, 32, 0, stream>>>(h, P(0), P(1), xn, DIMC, flip, 1e-5f);

    // proj_up: (784,192) @ (192->768)^T + b
    wmma_gemm_lds<<<dim3(7, (2 * INNER) / 16), GTHREADS, 0, stream>>>(
        xn, P(2), P(3), xi, 2 * INNER, DIMC, DIMC, 2 * INNER);

    dwconv_silu_kernel<<<(BS * INNER + T - 1) / T, T, 0, stream>>>(xi, P(7), P(8), xa);

    qkv_kernel<<<(BS * INNER + T - 1) / T, T, 0, stream>>>(xa, xi, P(4), P(5), P(6), gin);

    // gates: (784,1152) @ (1152->96)^T + b
    wmma_gemm_lds<<<dim3(7, NH / 16), GTHREADS, 0, stream>>>(
        gin, P(9), P(10), igb, NH, GIN_LD, GIN_LD, NH);
    wmma_gemm_lds<<<dim3(7, NH / 16), GTHREADS, 0, stream>>>(
        gin, P(11), P(12), fgb, NH, GIN_LD, GIN_LD, NH);

    mlstm_kernel<<<BATCH * NH, 256, 0, stream>>>(gin, igb, fgb, P(13), P(14), hout);

    gate_kernel<<<(BS * INNER + T - 1) / T, T, 0, stream>>>(hout, xa, xi, P(15));

    // proj_down: (784,384) @ (384->192)^T + b
    wmma_gemm_lds<<<dim3(7, DIMC / 16), GTHREADS, 0, stream>>>(
        hout, P(16), P(17), tmp, DIMC, INNER, INNER, DIMC);

    resadd_kernel<<<(BS * DIMC + T - 1) / T, T, 0, stream>>>(h, tmp, flip);
  }

  ln_kernel<<<BS, 32, 0, stream>>>(h, final_nw, final_nb, xn, DIMC, 0, 1e-6f);
  head_kernel<<<(BATCH * NCLS + T - 1) / T, T, 0, stream>>>(xn, head_w, head_b, (float*)d_out);
}